// Rconv_3D_Down_5205500363258
// MI455X (gfx1250) — compile-verified
//
#include <hip/hip_runtime.h>

typedef __attribute__((ext_vector_type(2))) float v2f;
typedef __attribute__((ext_vector_type(8))) float v8f;

// Problem constants (B=2, DIM=64, DIM2=128, HEADS=4, S=32)
#define NB     2
#define CIN    64
#define C3     192     // 3*DIM
#define NVOX   32768   // 32^3
#define LQ     4096    // 16^3
#define NOUTC  128

// ---------------------------------------------------------------------------
// Pointwise (1x1x1 conv) GEMM via V_WMMA_F32_16X16X4_F32, K = 64.
//   Y[b, m, n] = bias[m] + sum_{k<64} W[m,k] * X[b, k, n]
// Compile-time NCOLS -> all X/W loads become immediate-offset loads from one
// base address. Each wave computes MB consecutive 16x16 M-tiles for one
// N-tile, reusing the B operand MB times (64 v_wmma per wave at MB=4).
// A layout (16x4 f32): lanes 0-15 hold M=lane&15, VGPR j holds K=j+2*(lane>>4)
// B layout (4x16 f32): VGPR j, N=lane&15, K=j+2*(lane>>4)
// C/D layout: VGPR r -> row r + 8*(lane>>4), N = lane&15
// ---------------------------------------------------------------------------
template<int NCOLS, int MROWS, int MB>
__global__ __launch_bounds__(256) void gemm_pointwise_wmma(
    const float* __restrict__ W, const float* __restrict__ bias,
    const float* __restrict__ X, float* __restrict__ Y)
{
    constexpr int MG = (MROWS / 16) / MB;   // m-tile groups
    constexpr int NT = NCOLS / 16;          // n tiles
    int wave = (blockIdx.x * blockDim.x + threadIdx.x) >> 5;
    int lane = threadIdx.x & 31;
    int vt   = wave % NT;
    int rest = wave / NT;
    int mg   = rest % MG;
    int b    = rest / MG;
    if (b >= NB) return;                    // wave-uniform; grid sized exactly

    int lo = lane & 15, hi = lane >> 4;
    int col = vt * 16 + lo;
    const float* Xb = X + (size_t)b * CIN * NCOLS + col;   // column base
    const float* Wb = W + (mg * MB * 16 + lo) * 64;        // row base

    v8f acc[MB];
    #pragma unroll
    for (int i = 0; i < MB; ++i)
        #pragma unroll
        for (int r = 0; r < 8; ++r) acc[i][r] = 0.f;

    #pragma unroll
    for (int kk = 0; kk < 16; ++kk) {
        int k0 = kk * 4 + hi * 2;
        v2f bb;
        bb.x = Xb[(size_t)k0 * NCOLS];
        bb.y = Xb[(size_t)(k0 + 1) * NCOLS];
        #pragma unroll
        for (int i = 0; i < MB; ++i) {
            v2f a = *(const v2f*)(Wb + i * 16 * 64 + k0);
            acc[i] = __builtin_amdgcn_wmma_f32_16x16x4_f32(
                         false, a, false, bb, (short)0, acc[i], false, false);
        }
    }

    float* Yb = Y + (size_t)b * MROWS * NCOLS + col;
    #pragma unroll
    for (int i = 0; i < MB; ++i) {
        #pragma unroll
        for (int r = 0; r < 8; ++r) {
            int row = mg * MB * 16 + i * 16 + r + hi * 8;
            Yb[(size_t)row * NCOLS] = acc[i][r] + bias[row];
        }
    }
}

// ---------------------------------------------------------------------------
// Depthwise 3x3x3 conv, zero padding, 192 channels.
// One thread -> 4 consecutive w-voxels; 6-wide sliding row window per tap-row
// (aligned float4 + 2 edge scalars) -> ~13.5 loads/output, b128 stores.
// ---------------------------------------------------------------------------
__global__ __launch_bounds__(256) void dwconv3(
    const float* __restrict__ Yin, const float* __restrict__ W2,
    const float* __restrict__ B2, float* __restrict__ Yout)
{
    int idx = blockIdx.x * 256 + threadIdx.x;   // (b*192+ch)*8192 + quad
    int v4  = idx & 8191;
    int bc  = idx >> 13;
    int ch  = bc % C3;
    int s   = v4 >> 8;
    int hh  = (v4 >> 3) & 31;
    int w0  = (v4 & 7) * 4;
    const float* base = Yin + (size_t)bc * NVOX;
    const float* w = W2 + ch * 27;
    float bv = B2[ch];
    float acc0 = bv, acc1 = bv, acc2 = bv, acc3 = bv;
    #pragma unroll
    for (int di = 0; di < 3; ++di) {
        int zs = s + di - 1;
        if (zs < 0 || zs > 31) continue;
        #pragma unroll
        for (int dj = 0; dj < 3; ++dj) {
            int zh = hh + dj - 1;
            if (zh < 0 || zh > 31) continue;
            const float* row = base + zs * 1024 + zh * 32 + w0;
            float4 mid = *(const float4*)row;
            float left  = (w0 > 0)  ? row[-1] : 0.f;
            float right = (w0 < 28) ? row[4]  : 0.f;
            float w_0 = w[di*9 + dj*3 + 0];
            float w_1 = w[di*9 + dj*3 + 1];
            float w_2 = w[di*9 + dj*3 + 2];
            acc0 += w_0*left  + w_1*mid.x + w_2*mid.y;
            acc1 += w_0*mid.x + w_1*mid.y + w_2*mid.z;
            acc2 += w_0*mid.y + w_1*mid.z + w_2*mid.w;
            acc3 += w_0*mid.z + w_1*mid.w + w_2*right;
        }
    }
    float4 res; res.x = acc0; res.y = acc1; res.z = acc2; res.w = acc3;
    *(float4*)(Yout + (size_t)bc * NVOX + s * 1024 + hh * 32 + w0) = res;
}

// ---------------------------------------------------------------------------
// Region sums S[which,b,h,kk,cc] = sum over depth slab [8h,8h+8) x 32 x 32 of
// edge-clamped neighbor (kk) of channel cc of k (which=0) / v (which=1).
// Block = 8 waves; each wave owns one channel, lanes sweep ww (coalesced;
// zw = clamp(lane+dk) is loop-invariant, rest of address is SALU-uniform).
// ---------------------------------------------------------------------------
__global__ __launch_bounds__(256) void region_sum(
    const float* __restrict__ Q2, float* __restrict__ Sout)
{
    int bi = blockIdx.x;                 // 3456 = 2*2*4*27*8
    int oct = bi & 7; bi >>= 3;
    int kk = bi % 27; bi /= 27;
    int h = bi & 3;  bi >>= 2;
    int b = bi & 1;  int which = bi >> 1;
    int wv   = threadIdx.x >> 5;
    int lane = threadIdx.x & 31;
    int cc   = oct * 8 + wv;
    int di = kk / 9 - 1, dj = (kk / 3) % 3 - 1, dk = kk % 3 - 1;
    int ch = 64 + which * 64 + cc;       // k: [64,128) v: [128,192)
    const float* base = Q2 + (size_t)(b * C3 + ch) * NVOX;

    int zw = min(max(lane + dk, 0), 31);
    float acc = 0.f;
    for (int so = 0; so < 8; ++so) {
        int zs = min(max(h * 8 + so + di, 0), 31);
        const float* p0 = base + zs * 1024 + zw;
        #pragma unroll 4
        for (int hh = 0; hh < 32; ++hh) {
            int zh = min(max(hh + dj, 0), 31);
            acc += p0[zh * 32];
        }
    }
    #pragma unroll
    for (int off = 16; off > 0; off >>= 1)
        acc += __shfl_xor(acc, off, 32);
    if (lane == 0)
        Sout[((((size_t)which * NB + b) * 4 + h) * 27 + kk) * 64 + cc] = acc;
}

// ---------------------------------------------------------------------------
// Fold 1728-vector into 432: KV[which,b,h,j] = sum_r S[.., r*432+j]
// (S's [kk][cc] flattening is exactly p = kk*64+cc)
// ---------------------------------------------------------------------------
__global__ __launch_bounds__(256) void fold_sums(
    const float* __restrict__ Sin, float* __restrict__ KV)
{
    int idx = blockIdx.x * 256 + threadIdx.x;     // 2*2*4*432 = 6912 exactly
    int j = idx % 432; int t = idx / 432;
    int h = t & 3; t >>= 2; int b = t & 1; int which = t >> 1;
    const float* Sp = Sin + (((size_t)which * NB + b) * 4 + h) * 27 * 64;
    float acc = 0.f;
    #pragma unroll
    for (int r = 0; r < 4; ++r) acc += Sp[r * 432 + j];
    KV[(((size_t)which * NB + b) * 4 + h) * 432 + j] = acc;
}

// ---------------------------------------------------------------------------
// Attention: one thread per (b,h,q). 27-way softmax against the per-(b,h)
// ksum/vsum tables (27x16) held in LDS; output in (B, C=64, 16^3) order.
// q maps to qn[b, l=h*1024+q/4, c=(q%4)*16+c']; qn samples q-tensor at ::2.
// ---------------------------------------------------------------------------
__global__ __launch_bounds__(256) void attn_kernel(
    const float* __restrict__ Q2, const float* __restrict__ KV,
    float* __restrict__ Osp)
{
    int bh  = blockIdx.x >> 4;          // (b*4+h)
    int qbk = blockIdx.x & 15;
    int b = bh >> 2; int h = bh & 3;

    __shared__ float ks[432];
    __shared__ float vs[432];
    const float* kp = KV + ((size_t)(0 * NB + b) * 4 + h) * 432;
    const float* vp = KV + ((size_t)(1 * NB + b) * 4 + h) * 432;
    for (int i = threadIdx.x; i < 432; i += 256) { ks[i] = kp[i]; vs[i] = vp[i]; }
    __syncthreads();

    int q = qbk * 256 + threadIdx.x;
    int m = q >> 2, r = q & 3;
    int l = h * 1024 + m;
    int sd = l >> 8, hd = (l >> 4) & 15, wd = l & 15;
    int vox = (2 * sd) * 1024 + (2 * hd) * 32 + (2 * wd);

    float qv[16];
    #pragma unroll
    for (int c = 0; c < 16; ++c)
        qv[c] = Q2[(size_t)(b * C3 + (r * 16 + c)) * NVOX + vox];

    float logits[27]; float mx = -1e30f;
    #pragma unroll
    for (int k = 0; k < 27; ++k) {
        float d = 0.f;
        #pragma unroll
        for (int c = 0; c < 16; ++c) d += qv[c] * ks[k * 16 + c];
        d *= 0.25f;                                  // (C/heads)^-0.5
        logits[k] = d; mx = fmaxf(mx, d);
    }
    float sum = 0.f;
    #pragma unroll
    for (int k = 0; k < 27; ++k) { logits[k] = __expf(logits[k] - mx); sum += logits[k]; }
    float inv = 1.f / sum;
    #pragma unroll
    for (int c = 0; c < 16; ++c) {
        float o = 0.f;
        #pragma unroll
        for (int k = 0; k < 27; ++k) o += logits[k] * vs[k * 16 + c];
        Osp[(size_t)(b * CIN + (r * 16 + c)) * LQ + l] = o * inv;
    }
}

// ---------------------------------------------------------------------------
extern "C" void kernel_launch(void* const* d_in, const int* in_sizes, int n_in,
                              void* d_out, int out_size, void* d_ws, size_t ws_size,
                              hipStream_t stream)
{
    const float* x  = (const float*)d_in[0];
    const float* w1 = (const float*)d_in[1];
    const float* b1 = (const float*)d_in[2];
    const float* w2 = (const float*)d_in[3];
    const float* b2 = (const float*)d_in[4];
    const float* pw = (const float*)d_in[5];
    const float* pb = (const float*)d_in[6];
    float* out = (float*)d_out;

    char* ws = (char*)d_ws;
    size_t off = 0;
    float* y1   = (float*)(ws + off); off += (size_t)NB * C3 * NVOX * 4;      // 50.3 MB
    float* y2   = (float*)(ws + off); off += (size_t)NB * C3 * NVOX * 4;      // 50.3 MB
    float* Ssum = (float*)(ws + off); off += (size_t)2 * NB * 4 * 27 * 64 * 4;
    float* KV   = (float*)(ws + off); off += (size_t)2 * NB * 4 * 432 * 4;
    float* osp  = (float*)(ws + off); off += (size_t)NB * CIN * LQ * 4;       // 2 MB

    // 1) qkv1 1x1x1 conv: waves = 2 * (12/4) * 2048 = 12288 -> 1536 blocks
    gemm_pointwise_wmma<NVOX, C3, 4><<<1536, 256, 0, stream>>>(w1, b1, x, y1);
    // 2) depthwise 3x3x3: 2*192*8192 quad-threads -> 12288 blocks
    dwconv3<<<12288, 256, 0, stream>>>(y1, w2, b2, y2);
    // 3) region sums: 2*2*4*27*8 = 3456 blocks
    region_sum<<<3456, 256, 0, stream>>>(y2, Ssum);
    // 4) fold: 6912 threads
    fold_sums<<<27, 256, 0, stream>>>(Ssum, KV);
    // 5) attention: (2*4)*16 = 128 blocks
    attn_kernel<<<128, 256, 0, stream>>>(y2, KV, osp);
    // 6) proj GEMM: waves = 2 * (8/4) * 256 = 1024 -> 128 blocks
    gemm_pointwise_wmma<LQ, NOUTC, 4><<<128, 256, 0, stream>>>(pw, pb, osp, out);
}